// deform_dss_48996986913002
// MI455X (gfx1250) — compile-verified
//
#include <hip/hip_runtime.h>

// ---------------- problem constants ----------------
// B=64, N=64, S=4, NUM_CLS=2 (deform groups), H=W=28, K=3
#define FF 256          // B*S flattened image count
#define EPSV 1e-5f

typedef __attribute__((ext_vector_type(2))) float v2f;
typedef __attribute__((ext_vector_type(8))) float v8f;

// Full-precision CDNA5 matrix op: D(16x16,f32) += A(16x4,f32) * B(4x16,f32)
__device__ __forceinline__ v8f wmma_f32_k4(v2f a, v2f b, v8f c) {
  return __builtin_amdgcn_wmma_f32_16x16x4_f32(false, a, false, b, (short)0, c,
                                               false, false);
}

// LDS pair-interleaved im2col layout: element (k,n) lives at
//   cols[(k>>1)*32 + n*2 + (k&1)]
// so a GEMM lane reads B-fragment (k0+half*2, k0+half*2+1) as one aligned b64.
__device__ __forceinline__ int colIdx(int k, int n) {
  return ((k >> 1) << 5) + (n << 1) + (k & 1);
}

// ---------------------------------------------------------------------------
// Stage 1: conv 1->64, 3x3, pad 1, +bias, ReLU, deterministic partial stats.
// grid(FF, 64), block 128.  psum/psq[(c*4+s)*64 + b]
// ---------------------------------------------------------------------------
__global__ __launch_bounds__(128)
void k_conv1(const float* __restrict__ x, const float* __restrict__ w,
             const float* __restrict__ bias, float* __restrict__ h,
             float* __restrict__ psum, float* __restrict__ psq) {
  const int f = blockIdx.x, c = blockIdx.y;
  const int b = f >> 2, s = f & 3;
  float wr[9];
#pragma unroll
  for (int i = 0; i < 9; ++i) wr[i] = w[c * 9 + i];
  const float bi = bias[c];
  const float* xi = x + f * 784;
  float ls = 0.f, lq = 0.f;
  for (int p = threadIdx.x; p < 784; p += 128) {
    const int py = p / 28, px = p % 28;
    float acc = bi;
#pragma unroll
    for (int r = 0; r < 3; ++r)
#pragma unroll
      for (int t = 0; t < 3; ++t) {
        const int yy = py + r - 1, xx = px + t - 1;
        if (yy >= 0 && yy < 28 && xx >= 0 && xx < 28)
          acc += xi[yy * 28 + xx] * wr[r * 3 + t];
      }
    acc = fmaxf(acc, 0.f);
    h[(f * 64 + c) * 784 + p] = acc;
    ls += acc;
    lq += acc * acc;
  }
  __shared__ float sh[128];
  sh[threadIdx.x] = ls; __syncthreads();
  for (int o = 64; o > 0; o >>= 1) {
    if (threadIdx.x < o) sh[threadIdx.x] += sh[threadIdx.x + o];
    __syncthreads();
  }
  if (threadIdx.x == 0) psum[(c * 4 + s) * 64 + b] = sh[0];
  __syncthreads();
  sh[threadIdx.x] = lq; __syncthreads();
  for (int o = 64; o > 0; o >>= 1) {
    if (threadIdx.x < o) sh[threadIdx.x] += sh[threadIdx.x + o];
    __syncthreads();
  }
  if (threadIdx.x == 0) psq[(c * 4 + s) * 64 + b] = sh[0];
}

// ---------------------------------------------------------------------------
// Elementwise (optional in-place ReLU) + deterministic partial stats.
// grid(FF, C), block 128.
// ---------------------------------------------------------------------------
__global__ __launch_bounds__(128)
void k_stats(float* __restrict__ h, float* __restrict__ psum,
             float* __restrict__ psq, int C, int P, int do_relu) {
  const int f = blockIdx.x, c = blockIdx.y;
  const int b = f >> 2, s = f & 3;
  float* hp = h + (size_t)(f * C + c) * P;
  float ls = 0.f, lq = 0.f;
  for (int p = threadIdx.x; p < P; p += 128) {
    float v = hp[p];
    if (do_relu) { v = fmaxf(v, 0.f); hp[p] = v; }
    ls += v;
    lq += v * v;
  }
  __shared__ float sh[128];
  sh[threadIdx.x] = ls; __syncthreads();
  for (int o = 64; o > 0; o >>= 1) {
    if (threadIdx.x < o) sh[threadIdx.x] += sh[threadIdx.x + o];
    __syncthreads();
  }
  if (threadIdx.x == 0) psum[(c * 4 + s) * 64 + b] = sh[0];
  __syncthreads();
  sh[threadIdx.x] = lq; __syncthreads();
  for (int o = 64; o > 0; o >>= 1) {
    if (threadIdx.x < o) sh[threadIdx.x] += sh[threadIdx.x + o];
    __syncthreads();
  }
  if (threadIdx.x == 0) psq[(c * 4 + s) * 64 + b] = sh[0];
}

// ---------------------------------------------------------------------------
// BN finalize: scale/shift per (c,s).  smode: 0 -> gamma[c], 1 -> gamma[s]
// ---------------------------------------------------------------------------
__global__ void k_bn_finalize(const float* __restrict__ psum,
                              const float* __restrict__ psq,
                              const float* __restrict__ g,
                              const float* __restrict__ be,
                              float* __restrict__ scale,
                              float* __restrict__ shift, int C, float inv_cnt,
                              int smode) {
  const int idx = blockIdx.x * blockDim.x + threadIdx.x;
  if (idx >= C * 4) return;
  const int c = idx >> 2, s = idx & 3;
  float su = 0.f, sq = 0.f;
  for (int b = 0; b < 64; ++b) {
    su += psum[idx * 64 + b];
    sq += psq[idx * 64 + b];
  }
  const float mean = su * inv_cnt;
  const float var = sq * inv_cnt - mean * mean;
  const float gg = smode ? g[s] : g[c];
  const float bb = smode ? be[s] : be[c];
  const float sc = gg * rsqrtf(var + EPSV);
  scale[idx] = sc;
  shift[idx] = bb - mean * sc;
}

// In-place BN apply (stage 1 only).
__global__ void k_bn_apply(float* __restrict__ h, const float* __restrict__ scale,
                           const float* __restrict__ shift, int C, int P,
                           int total) {
  const int idx = blockIdx.x * blockDim.x + threadIdx.x;
  if (idx >= total) return;
  const int fc = idx / P;
  const int c = fc % C, f = fc / C;
  const int e = c * 4 + (f & 3);
  h[idx] = h[idx] * scale[e] + shift[e];
}

// ---------------------------------------------------------------------------
// Standard 3x3 conv (offset branches) as LDS-staged implicit GEMM with fp32
// WMMA.  grid(FF, ceil(P/16)), block 128 (4 waves).
// Stage: 144 (kk,n) tuples compute mask+src offset once, then sweep CIN.
// GEMM: wave w handles M-tile mt=w (COUT=36 -> 3 tiles, padded reads clamped,
// stores guarded).  All loads unconditional; B-frag = 1 ds b64, A = 1 b64.
// ---------------------------------------------------------------------------
template <int CIN, int COUT, int HW>
__global__ __launch_bounds__(128)
void k_conv3x3_wmma(const float* __restrict__ in, const float* __restrict__ wt,
                    const float* __restrict__ bias, float* __restrict__ out) {
  constexpr int P = HW * HW;
  constexpr int K = CIN * 9;
  constexpr int MT = (COUT + 15) / 16;
  __shared__ float cols[K * 16];

  const int f = blockIdx.x, nt = blockIdx.y;
  const int tid = threadIdx.x;
  const float* inf = in + (size_t)f * CIN * P;

  // ---- stage im2col tile ----
  for (int t = tid; t < 144; t += 128) {
    const int n = t & 15;
    const int kk = t >> 4;            // 0..8
    const int r = kk / 3, tt = kk % 3;
    const int p = nt * 16 + n;
    const bool pv = p < P;
    const int pc = pv ? p : 0;
    const int py = pc / HW, px = pc % HW;
    const int yy = py + r - 1, xx = px + tt - 1;
    const bool ok = pv && yy >= 0 && yy < HW && xx >= 0 && xx < HW;
    const float maskf = ok ? 1.f : 0.f;
    const int src = ok ? (yy * HW + xx) : 0;
    int k = kk;
    const float* ip = inf + src;
#pragma unroll 4
    for (int ci = 0; ci < CIN; ++ci) {
      cols[colIdx(k, n)] = maskf * ip[0];
      ip += P;
      k += 9;
    }
  }
  __syncthreads();

  // ---- WMMA GEMM ----
  const int lane = tid & 31, wave = tid >> 5;
  const int l = lane & 15, half = lane >> 4;
  const int p = nt * 16 + l;
  for (int mt = wave; mt < MT; mt += 4) {
    const int m = mt * 16 + l;
    const int mcl = (m < COUT) ? m : (COUT - 1);   // clamp; bad rows not stored
    const v2f* wrow = (const v2f*)(wt + (size_t)mcl * K);
    v8f c = {};
#pragma unroll 4
    for (int k0 = 0; k0 < K; k0 += 4) {
      const int kh = (k0 >> 1) + half;
      const v2f a = wrow[kh];
      const v2f bfrag = *(const v2f*)(&cols[(kh << 5) + (l << 1)]);
      c = wmma_f32_k4(a, bfrag, c);
    }
    if (p < P) {
#pragma unroll
      for (int j = 0; j < 8; ++j) {
        const int mo = mt * 16 + half * 8 + j;
        if (mo < COUT) out[((size_t)f * COUT + mo) * P + p] = c[j] + bias[mo];
      }
    }
  }
}

// ---------------------------------------------------------------------------
// Deformable 3x3 conv (2 groups), LDS-staged implicit GEMM with fp32 WMMA.
// grid(FF, ceil(P/16)), block 256 (8 waves).
// Stage A: preload 36x16 offset tile to LDS.
// Stage B: 288 (g,kk,n) tuples compute 4 bilinear taps ONCE, sweep Cg chans.
// GEMM: wave handles M-tiles mt = wave, wave+8 ...
// ---------------------------------------------------------------------------
template <int CIN, int COUT, int HW>
__global__ __launch_bounds__(256)
void k_deform_wmma(const float* __restrict__ in, const float* __restrict__ off,
                   const float* __restrict__ wt, const float* __restrict__ bias,
                   float* __restrict__ out) {
  constexpr int P = HW * HW;
  constexpr int K = CIN * 9;
  constexpr int Cg = CIN / 2;
  __shared__ float cols[K * 16];
  __shared__ float offs[36 * 16];

  const int f = blockIdx.x, nt = blockIdx.y;
  const int tid = threadIdx.x;
  const float* inf = in + (size_t)f * CIN * P;
  const float* offf = off + (size_t)f * 36 * P;

  // ---- preload offset tile (clamped-address unconditional loads) ----
  for (int e = tid; e < 36 * 16; e += 256) {
    const int ch = e >> 4, n = e & 15;
    const int p = nt * 16 + n;
    const int pc = (p < P) ? p : 0;
    offs[e] = offf[ch * P + pc];
  }
  __syncthreads();

  // ---- stage bilinear im2col tile: tuple = (g, kk, n) ----
  for (int t = tid; t < 2 * 9 * 16; t += 256) {
    const int n = t & 15;
    const int gk = t >> 4;            // g*9 + kk, 0..17
    const int g = gk / 9, kk = gk % 9;
    const int ki = kk / 3, kj = kk % 3;
    const int p = nt * 16 + n;
    const float pvf = (p < P) ? 1.f : 0.f;
    const int pc = (p < P) ? p : 0;
    const int py = pc / HW, px = pc % HW;
    const float oy = offs[((g * 9 + kk) * 2 + 0) * 16 + n];
    const float ox = offs[((g * 9 + kk) * 2 + 1) * 16 + n];
    const float y = oy + (float)(py - 1 + ki);
    const float x = ox + (float)(px - 1 + kj);
    const float y0f = floorf(y), x0f = floorf(x);
    const float wy = y - y0f, wx = x - x0f;
    const float hwm1 = (float)(HW - 1);
    const float vy0 = (y0f >= 0.f && y0f <= hwm1) ? 1.f : 0.f;
    const float vy1 = (y0f + 1.f >= 0.f && y0f + 1.f <= hwm1) ? 1.f : 0.f;
    const float vx0 = (x0f >= 0.f && x0f <= hwm1) ? 1.f : 0.f;
    const float vx1 = (x0f + 1.f >= 0.f && x0f + 1.f <= hwm1) ? 1.f : 0.f;
    const int y0 = (int)fminf(fmaxf(y0f, 0.f), hwm1);
    const int x0 = (int)fminf(fmaxf(x0f, 0.f), hwm1);
    const int y1 = (int)fminf(fmaxf(y0f + 1.f, 0.f), hwm1);
    const int x1 = (int)fminf(fmaxf(x0f + 1.f, 0.f), hwm1);
    const float w00 = pvf * vy0 * vx0 * (1.f - wy) * (1.f - wx);
    const float w01 = pvf * vy0 * vx1 * (1.f - wy) * wx;
    const float w10 = pvf * vy1 * vx0 * wy * (1.f - wx);
    const float w11 = pvf * vy1 * vx1 * wy * wx;
    const int i00 = y0 * HW + x0, i01 = y0 * HW + x1;
    const int i10 = y1 * HW + x0, i11 = y1 * HW + x1;
    const float* img = inf + g * Cg * P;
    int k = g * Cg * 9 + kk;
#pragma unroll 4
    for (int ci = 0; ci < Cg; ++ci) {
      const float v = w00 * img[i00] + w01 * img[i01] +
                      w10 * img[i10] + w11 * img[i11];
      cols[colIdx(k, n)] = v;
      img += P;
      k += 9;
    }
  }
  __syncthreads();

  // ---- WMMA GEMM ----
  const int lane = tid & 31, wave = tid >> 5;
  const int l = lane & 15, half = lane >> 4;
  const int p = nt * 16 + l;
  for (int mt = wave; mt * 16 < COUT; mt += 8) {
    const int m = mt * 16 + l;                    // COUT % 16 == 0
    const v2f* wrow = (const v2f*)(wt + (size_t)m * K);
    v8f c = {};
#pragma unroll 4
    for (int k0 = 0; k0 < K; k0 += 4) {
      const int kh = (k0 >> 1) + half;
      const v2f a = wrow[kh];
      const v2f bfrag = *(const v2f*)(&cols[(kh << 5) + (l << 1)]);
      c = wmma_f32_k4(a, bfrag, c);
    }
    if (p < P) {
#pragma unroll
      for (int j = 0; j < 8; ++j) {
        const int mo = mt * 16 + half * 8 + j;
        out[((size_t)f * COUT + mo) * P + p] = c[j] + bias[mo];
      }
    }
  }
}

// ---------------------------------------------------------------------------
// Fused BN apply + 2x2 maxpool.  emb_mode: 0 -> (f,c,Q,Q) ws layout,
// 1 -> reference embedding layout (b, c, s, Q, Q) into d_out.
// grid(FF, C), block 128.
// ---------------------------------------------------------------------------
__global__ __launch_bounds__(128)
void k_bn_pool(const float* __restrict__ in, const float* __restrict__ scale,
               const float* __restrict__ shift, float* __restrict__ out, int C,
               int HWin, int emb_mode) {
  const int f = blockIdx.x, c = blockIdx.y;
  const int b = f >> 2, s = f & 3;
  const int Q = HWin >> 1;
  const float sc = scale[c * 4 + s], sf = shift[c * 4 + s];
  const float* ip = in + (size_t)(f * C + c) * HWin * HWin;
  float* op = emb_mode ? out + ((size_t)(b * C + c) * 4 + s) * Q * Q
                       : out + (size_t)(f * C + c) * Q * Q;
  for (int q = threadIdx.x; q < Q * Q; q += 128) {
    const int qy = q / Q, qx = q % Q;
    const float* r0 = ip + (2 * qy) * HWin + 2 * qx;
    const float a0 = r0[0] * sc + sf;
    const float a1 = r0[1] * sc + sf;
    const float a2 = r0[HWin] * sc + sf;
    const float a3 = r0[HWin + 1] * sc + sf;
    op[q] = fmaxf(fmaxf(a0, a1), fmaxf(a2, a3));
  }
}

// ---------------------------------------------------------------------------
// Head: pooled = relu(max over S of embedding); logits = pooled @ fc_w^T + fc_b
// grid(64), block 256.
// ---------------------------------------------------------------------------
__global__ __launch_bounds__(256)
void k_fc(const float* __restrict__ emb, const float* __restrict__ fw,
          const float* __restrict__ fb, float* __restrict__ logits) {
  const int b = blockIdx.x;
  const int tid = threadIdx.x;
  float acc[10];
#pragma unroll
  for (int o = 0; o < 10; ++o) acc[o] = 0.f;
  for (int i = tid; i < 12544; i += 256) {
    const int c = i / 49, p = i % 49;
    const float* e = emb + ((size_t)(b * 256 + c) * 4) * 49 + p;
    float m = fmaxf(fmaxf(e[0], e[49]), fmaxf(e[98], e[147]));
    m = fmaxf(m, 0.f);
#pragma unroll
    for (int o = 0; o < 10; ++o) acc[o] += m * fw[o * 12544 + i];
  }
  __shared__ float sh[256];
  for (int o = 0; o < 10; ++o) {
    sh[tid] = acc[o];
    __syncthreads();
    for (int ofs = 128; ofs > 0; ofs >>= 1) {
      if (tid < ofs) sh[tid] += sh[tid + ofs];
      __syncthreads();
    }
    if (tid == 0) logits[b * 10 + o] = sh[0] + fb[o];
    __syncthreads();
  }
}

// ---------------------------------------------------------------------------
extern "C" void kernel_launch(void* const* d_in, const int* in_sizes, int n_in,
                              void* d_out, int out_size, void* d_ws,
                              size_t ws_size, hipStream_t stream) {
  (void)in_sizes; (void)n_in; (void)out_size; (void)ws_size;
  const float* x    = (const float*)d_in[0];
  const float* c1w  = (const float*)d_in[1];
  const float* c1b  = (const float*)d_in[2];
  const float* o1w  = (const float*)d_in[3];
  const float* o1b  = (const float*)d_in[4];
  const float* d1w  = (const float*)d_in[5];
  const float* d1b  = (const float*)d_in[6];
  const float* o2w  = (const float*)d_in[7];
  const float* o2b  = (const float*)d_in[8];
  const float* d2w  = (const float*)d_in[9];
  const float* d2b  = (const float*)d_in[10];
  const float* bn1g = (const float*)d_in[11];
  const float* bn1b = (const float*)d_in[12];
  const float* bn2g = (const float*)d_in[13];
  const float* bn2b = (const float*)d_in[14];
  const float* bn3g = (const float*)d_in[15];
  const float* bn3b = (const float*)d_in[16];
  const float* fcw  = (const float*)d_in[17];
  const float* fcb  = (const float*)d_in[18];
  float* out = (float*)d_out;
  float* ws  = (float*)d_ws;

  // workspace layout (floats); total ~45.9M floats (~184 MB), L2-resident
  float* bufA  = ws;                    // h1 (256*64*784)  -> later pooled1
  float* bufB  = ws + 12845056;         // off1 (256*36*784) -> later off2
  float* bufC  = ws + 20070400;         // d1out (256*128*784) -> later d2out
  float* psum  = ws + 45760512;         // 1024*64
  float* psq   = psum + 65536;
  float* scale = psq + 65536;
  float* shift = scale + 1024;

  // stage 1: conv1 + relu + BN1
  k_conv1<<<dim3(FF, 64), 128, 0, stream>>>(x, c1w, c1b, bufA, psum, psq);
  k_bn_finalize<<<1, 256, 0, stream>>>(psum, psq, bn1g, bn1b, scale, shift, 64,
                                       1.f / 50176.f, 0);
  {
    const int total = FF * 64 * 784;
    k_bn_apply<<<(total + 255) / 256, 256, 0, stream>>>(bufA, scale, shift, 64,
                                                        784, total);
  }

  // stage 2: offset1 conv + deform1 + relu + BN2 + pool
  k_conv3x3_wmma<64, 36, 28><<<dim3(FF, 49), 128, 0, stream>>>(bufA, o1w, o1b,
                                                               bufB);
  k_deform_wmma<64, 128, 28><<<dim3(FF, 49), 256, 0, stream>>>(bufA, bufB, d1w,
                                                               d1b, bufC);
  k_stats<<<dim3(FF, 128), 128, 0, stream>>>(bufC, psum, psq, 128, 784, 1);
  k_bn_finalize<<<2, 256, 0, stream>>>(psum, psq, bn2g, bn2b, scale, shift, 128,
                                       1.f / 50176.f, 1);
  k_bn_pool<<<dim3(FF, 128), 128, 0, stream>>>(bufC, scale, shift, bufA, 128,
                                               28, 0);

  // stage 3: offset2 conv + deform2 + BN3 + pool -> embedding in d_out
  k_conv3x3_wmma<128, 36, 14><<<dim3(FF, 13), 128, 0, stream>>>(bufA, o2w, o2b,
                                                                bufB);
  k_deform_wmma<128, 256, 14><<<dim3(FF, 13), 256, 0, stream>>>(bufA, bufB, d2w,
                                                                d2b, bufC);
  k_stats<<<dim3(FF, 256), 128, 0, stream>>>(bufC, psum, psq, 256, 196, 0);
  k_bn_finalize<<<4, 256, 0, stream>>>(psum, psq, bn3g, bn3b, scale, shift, 256,
                                       1.f / 12544.f, 1);
  k_bn_pool<<<dim3(FF, 256), 128, 0, stream>>>(bufC, scale, shift, out + 640,
                                               256, 14, 1);

  // head: max-over-S + relu + FC
  k_fc<<<64, 256, 0, stream>>>(out + 640, fcw, fcb, out);
}